// STICKYKVCache_LayerWise_2405181686052
// MI455X (gfx1250) — compile-verified
//
#include <hip/hip_runtime.h>

// ---------------------------------------------------------------------------
// STICKYKVCache prefill bookkeeping for MI455X (gfx1250, wave32).
//
// Roofline: attn tensor = 1*32*2048*2048 f32 = 512 MiB >> 192 MB L2, so it
// streams from HBM exactly once: 512 MiB / 23.3 TB/s ~= 22 us. Everything
// else is < 20 MiB. The q-reduction (sum over queries) runs on the matrix
// pipe as ones(16x4) @ B(4x16) via V_WMMA_F32_16X16X4_F32.
//
// Key mapping trick: A == all-ones AND each window score sums all 32 of its
// kv columns. Therefore, for any hardware K<->(vgpr,half-lane) assignment and
// any column<->lane permutation, feeding a lane's contiguous float2
//   { data(row q0+half, col c+2n), data(row q0+half, col c+2n+1) }
// directly as the B operand makes the WMMA accumulate, per hardware column,
// exactly the 4 streamed values {2 rows}x{2 cols}; the final 16-lane
// reduction yields the exact window sum. So every global_load_b64 result is
// consumed by a WMMA with ZERO register shuffling (previous version needed
// v_dual_mov chains to cross-pair .x/.y) — VALU stays idle, loads stream.
// ---------------------------------------------------------------------------

typedef float v2f __attribute__((ext_vector_type(2)));
typedef float v8f __attribute__((ext_vector_type(8)));

#define NHEADS   32
#define SEQ      2048
#define SINK     4
#define OMEGA    32
#define NWIN     63          // (2048 - 4) / 32
#define MAXWIN   4096
#define QSPLIT   8           // q-chunks per (head, window) -> 16128 waves total
#define QCHUNK   (SEQ / QSPLIT)   // 256 rows per wave
#define LEDGER_ROWS 65536
#define LEDGER_COLS (2 + 2 * NHEADS)   // 66

// d_out layout (flat, return order): window_scores | local_history | token_ledger
#define WS_ELEMS  (NHEADS * MAXWIN * 3)   // 393216
#define LH_ELEMS  (NHEADS * MAXWIN)       // 131072
#define TL_OFF    (WS_ELEMS + LH_ELEMS)   // 524288

// ---- Kernel A: token ledger copy with prefill overrides --------------------
__global__ void sticky_ledger_kernel(const float* __restrict__ tl_in,
                                     float* __restrict__ tl_out, int n) {
    int idx = blockIdx.x * blockDim.x + threadIdx.x;
    if (idx >= n) return;
    int row = idx / LEDGER_COLS;
    int col = idx - row * LEDGER_COLS;
    float v = tl_in[idx];
    if (row < SEQ) {
        if (col == 0)                 v = (float)row;   // token id
        else if (col == 1)            v = 0.0f;         // LAYER_IDX
        else if (col < 2 + NHEADS)    v = (float)row;   // phys idx == id (prefill)
        // cols 34..65 keep input (-1)
    }
    tl_out[idx] = v;
}

// ---- Kernel B: bandwidth-bound q-reduction via WMMA ------------------------
// One wave per (head, window, q-chunk). Each iteration: two coalesced
// global_load_b64 (each = 2 rows x 32 cols = 256 B of stream) feeding two
// independent WMMA accumulators directly.
__global__ void __launch_bounds__(256)
sticky_reduce_kernel(const float* __restrict__ attn,
                     float* __restrict__ partials) {
    const int wave = (int)((blockIdx.x * blockDim.x + threadIdx.x) >> 5);
    const int lane = threadIdx.x & 31;
    const int qc   = wave & (QSPLIT - 1);
    const int t    = wave >> 3;            // QSPLIT == 8
    const int w    = t % NWIN;
    const int h    = t / NWIN;

    const int n    = lane & 15;
    const int half = lane >> 4;

    // Lane base: head h, q-chunk start row (+half), window col + 2*n.
    const float* base = attn + (size_t)h * SEQ * SEQ
                             + (size_t)(qc * QCHUNK + half) * SEQ
                             + (SINK + OMEGA * w) + 2 * n;

    const v2f a_ones = {1.0f, 1.0f};       // 16x4 all-ones A matrix
    v8f acc0 = {};
    v8f acc1 = {};

    #pragma unroll 8
    for (int q0 = 0; q0 < QCHUNK; q0 += 4) {
        const float* p0 = base + (size_t)q0 * SEQ;   // rows q0 / q0+1 (by half)
        const float* p1 = p0 + 2 * SEQ;              // rows q0+2 / q0+3
        v2f d0 = *(const v2f*)p0;          // global_load_b64, 128B per half-wave
        v2f d1 = *(const v2f*)p1;
        if (q0 + 16 < QCHUNK)              // stream ahead: global_prefetch_b8
            __builtin_prefetch(p0 + 16 * SEQ, 0, 0);

        // Loaded pairs feed WMMA B operands directly — no shuffles.
        acc0 = __builtin_amdgcn_wmma_f32_16x16x4_f32(
                   false, a_ones, false, d0, (short)0, acc0, false, false);
        acc1 = __builtin_amdgcn_wmma_f32_16x16x4_f32(
                   false, a_ones, false, d1, (short)0, acc1, false, false);
    }

    // D row 0: lane L holds the accumulated sum of its {2 rows}x{2 cols}
    // groups for hardware column L%16 (lanes 16-31 duplicate lanes 0-15).
    float s = acc0[0] + acc1[0];
    s += __shfl_xor(s, 1);
    s += __shfl_xor(s, 2);
    s += __shfl_xor(s, 4);
    s += __shfl_xor(s, 8);
    if (lane == 0)
        partials[(h * NWIN + w) * QSPLIT + qc] = s;   // deterministic (no atomics)
}

// ---- Kernel C: finalize window_scores + local_history ----------------------
__global__ void sticky_finalize_kernel(const float* __restrict__ ws_in,
                                       const float* __restrict__ partials,
                                       float* __restrict__ out, int n) {
    int idx = blockIdx.x * blockDim.x + threadIdx.x;
    if (idx >= n) return;
    if (idx < WS_ELEMS) {
        // window_scores[h][w][comp]
        int h    = idx / (MAXWIN * 3);
        int rem  = idx - h * (MAXWIN * 3);
        int w    = rem / 3;
        int comp = rem - w * 3;
        float v;
        if (w < NWIN) {
            if (comp == 0) {
                const float* p = partials + (h * NWIN + w) * QSPLIT;
                float s = 0.0f;
                #pragma unroll
                for (int i = 0; i < QSPLIT; ++i) s += p[i];
                v = s;
            } else {
                v = (float)w;              // comp 1 and 2 both = window index
            }
        } else {
            v = ws_in[idx];                // untouched entries keep input
        }
        out[idx] = v;
    } else {
        // local_history[h][w]: zeros except first NWIN windows
        int j = idx - WS_ELEMS;
        int h = j / MAXWIN;
        int w = j - h * MAXWIN;
        float v = 0.0f;
        if (w < NWIN) {
            const float* p = partials + (h * NWIN + w) * QSPLIT;
            #pragma unroll
            for (int i = 0; i < QSPLIT; ++i) v += p[i];
        }
        out[idx] = v;
    }
}

extern "C" void kernel_launch(void* const* d_in, const int* in_sizes, int n_in,
                              void* d_out, int out_size, void* d_ws, size_t ws_size,
                              hipStream_t stream) {
    const float* attn  = (const float*)d_in[0];  // [1,32,2048,2048]
    const float* tl_in = (const float*)d_in[1];  // [65536,66]
    const float* ws_in = (const float*)d_in[2];  // [32,4096,3]
    // d_in[3] (local_history) is fully overwritten by the reference -> unused.

    float* out      = (float*)d_out;
    float* partials = (float*)d_ws;              // 32*63*8 f32 = 63 KiB scratch

    // A: token ledger (independent of the reduction)
    {
        int n = LEDGER_ROWS * LEDGER_COLS;       // 4,325,376
        sticky_ledger_kernel<<<(n + 255) / 256, 256, 0, stream>>>(
            tl_in, out + TL_OFF, n);
    }
    // B: streaming WMMA reduction -> partials
    {
        int waves  = NHEADS * NWIN * QSPLIT;     // 16128 waves
        int blocks = waves / 8;                  // 8 waves (256 threads) per block
        sticky_reduce_kernel<<<blocks, 256, 0, stream>>>(attn, partials);
    }
    // C: finalize (stream order guarantees B has completed)
    {
        int n = WS_ELEMS + LH_ELEMS;             // 524,288
        sticky_finalize_kernel<<<(n + 255) / 256, 256, 0, stream>>>(
            ws_in, partials, out, n);
    }
}